// TP_nonlinearity_37374805410393
// MI455X (gfx1250) — compile-verified
//
#include <hip/hip_runtime.h>

typedef float v2f __attribute__((ext_vector_type(2)));
typedef float v8f __attribute__((ext_vector_type(8)));

#define NTRI 11

// Triples in setup_inputs()/_triples() order:
// (0,0,0)(0,1,1)(0,2,2)(1,1,0)(1,1,1)(1,1,2)(1,2,1)(1,2,2)(2,2,0)(2,2,1)(2,2,2)
__constant__ int c_l1[NTRI]  = {0,0,0,1,1,1,1,1,2,2,2};
__constant__ int c_l2[NTRI]  = {0,1,2,1,1,1,2,2,2,2,2};
__constant__ int c_l3[NTRI]  = {0,1,2,0,1,2,1,2,0,1,2};
// Position of each triple's 4096-channel block inside its l3 output
__constant__ int c_blk[NTRI] = {0,0,0,1,1,1,2,2,2,3,3};
// K padding shift per triple (Kpad = 4 for m<=3, 8 for m=5)
__constant__ int c_ksh[NTRI] = {2,2,2,2,2,2,2,2,3,3,3};
// LDS float offsets of padded T tables (size 64*Msz*Kpad each), with sentinel total
__constant__ int c_Toff[NTRI+1] = {0,256,1024,2304,2560,3328,4608,5376,6656,7168,8704,11264};
// LDS float offsets of w3j tables (size m*n*M each), with sentinel total
__constant__ int c_Woff[NTRI+1] = {0,1,10,35,44,71,116,161,236,261,336,461};
// Job prefix sums (jobs per triple = M*16 tiles), sentinel total = 560
__constant__ int c_Joff[NTRI+1] = {0,16,64,144,160,208,288,336,416,432,480,560};
// Padded x slice LDS offsets and K-stride shifts per l (strides 4,4,8)
__constant__ int c_xoff[3] = {0,256,512};
__constant__ int c_xsh[3]  = {2,2,3};
// Output geometry (floats): bases of out0/out1/out2 and per-batch strides
__constant__ long c_obase[3]   = {0L, 6291456L, 31457280L};
__constant__ int  c_bstride[3] = {12288, 49152, 81920};

#define XBASE 0            // 1024 floats of zero-padded x
#define TBASE 1024         // 11264 floats of zero-padded T
#define WBASE 12288        // 461 floats of w3j
#define LDS_FLOATS 12752

__global__ __launch_bounds__(256)
void tp_wmma_kernel(const float* __restrict__ x0,
                    const float* __restrict__ x1,
                    const float* __restrict__ x2,
                    const float* __restrict__ w0, const float* __restrict__ w1,
                    const float* __restrict__ w2, const float* __restrict__ w3,
                    const float* __restrict__ w4, const float* __restrict__ w5,
                    const float* __restrict__ w6, const float* __restrict__ w7,
                    const float* __restrict__ w8, const float* __restrict__ w9,
                    const float* __restrict__ w10,
                    float* __restrict__ out, int Btot)
{
    __shared__ float lds[LDS_FLOATS];
    const int b   = blockIdx.x;
    const int tid = threadIdx.x;

    // Prefetch next batch's largest x slice (gfx1250 global_prefetch_b8)
    if (b + 1 < Btot && tid == 0) {
        __builtin_prefetch(x2 + (size_t)(b + 1) * 320, 0, 0);
    }

    // ---- Phase 1a: stage x[l] into LDS, K-padded with zeros (1024 floats) ----
    // x0 -> [64][4], x1 -> [64][4], x2 -> [64][8]
    for (int idx = tid; idx < 1024; idx += 256) {
        int l, ch, k;
        if (idx < 512) { l = idx >> 8; int o = idx & 255; ch = o >> 2; k = o & 3; }
        else           { l = 2;        int o = idx - 512; ch = o >> 3; k = o & 7; }
        const int dim = 2 * l + 1;
        float v = 0.f;
        if (k < dim) {
            if (l == 0)      v = x0[(size_t)b *  64 + ch];
            else if (l == 1) v = x1[(size_t)b * 192 + ch * 3 + k];
            else             v = x2[(size_t)b * 320 + ch * 5 + k];
        }
        lds[XBASE + idx] = v;
    }

    // ---- Phase 1b: stage all w3j tables into LDS (461 floats) ----
    for (int idx = tid; idx < 461; idx += 256) {
        int t = 0;
        while (idx >= c_Woff[t + 1]) ++t;
        int o = idx - c_Woff[t];
        const float* wp;
        switch (t) {
            case 0:  wp = w0;  break; case 1:  wp = w1;  break;
            case 2:  wp = w2;  break; case 3:  wp = w3;  break;
            case 4:  wp = w4;  break; case 5:  wp = w5;  break;
            case 6:  wp = w6;  break; case 7:  wp = w7;  break;
            case 8:  wp = w8;  break; case 9:  wp = w9;  break;
            default: wp = w10; break;
        }
        lds[WBASE + idx] = wp[o];
    }
    __syncthreads();

    // ---- Phase 2: T[j][M][k] = sum_n w3j[k,n,M] * x_{l2}[j,n], zero for k >= m ----
    for (int idx = tid; idx < 11264; idx += 256) {
        int t = 0;
        while (idx >= c_Toff[t + 1]) ++t;
        const int rem = idx - c_Toff[t];
        const int m    = 2 * c_l1[t] + 1;
        const int n    = 2 * c_l2[t] + 1;
        const int Msz  = 2 * c_l3[t] + 1;
        const int ksh  = c_ksh[t];
        const int Kp   = 1 << ksh;
        const int MKp  = Msz << ksh;
        const int j    = rem / MKp;
        const int r2   = rem - j * MKp;
        const int Mv   = r2 >> ksh;
        const int kk   = r2 & (Kp - 1);
        float s = 0.f;
        if (kk < m) {
            const int xo2 = XBASE + c_xoff[c_l2[t]];
            const int sh2 = c_xsh[c_l2[t]];
            const int wo  = WBASE + c_Woff[t];
            for (int nn = 0; nn < n; ++nn)
                s += lds[wo + (kk * n + nn) * Msz + Mv] * lds[xo2 + (j << sh2) + nn];
        }
        lds[TBASE + idx] = s;
    }
    __syncthreads();

    // ---- Phase 3: WMMA jobs. job = (triple, M, 16x16 tile of the 64x64 (i,j) block)
    // Out_M[i,j] = X_{l1}(64 x m) @ T_M^T(m x 64); K padded with zeros -> branch-free.
    const int wv   = tid >> 5;        // wave id, 0..7
    const int lane = tid & 31;
    const int half = lane >> 4;       // 0: lanes 0-15, 1: lanes 16-31
    const int lr   = lane & 15;
    const int kA   = half ? 2 : 0;    // even -> 8B-aligned ds_load_b64 fragments

    for (int job = wv; job < 560; job += 8) {   // wave-uniform loop: EXEC all ones
        int t = 0;
        while (job >= c_Joff[t + 1]) ++t;
        const int rem  = job - c_Joff[t];
        const int Mv   = rem >> 4;
        const int tile = rem & 15;
        const int ti   = tile >> 2;
        const int tj   = tile & 3;
        const int m    = 2 * c_l1[t] + 1;
        const int Msz  = 2 * c_l3[t] + 1;
        const int ksh  = c_ksh[t];
        const int sh1  = c_xsh[c_l1[t]];
        const int aAddr = XBASE + c_xoff[c_l1[t]] + (((ti * 16 + lr) << sh1) + kA);
        const int bAddr = TBASE + c_Toff[t] + ((((tj * 16 + lr) * Msz + Mv) << ksh) + kA);

        v8f c = {0.f, 0.f, 0.f, 0.f, 0.f, 0.f, 0.f, 0.f};
        for (int kb = 0; kb < m; kb += 4) {     // 1 step for m<=4, 2 steps for m=5
            // 16x4 f32 A layout: VGPR0 = K{0,2}, VGPR1 = K{1,3}; pairs contiguous in LDS
            const v2f a  = *(const v2f*)(lds + aAddr + kb);
            const v2f bf = *(const v2f*)(lds + bAddr + kb);
            c = __builtin_amdgcn_wmma_f32_16x16x4_f32(
                    /*neg_a=*/false, a, /*neg_b=*/false, bf,
                    /*c_mod=*/(short)0, c, /*reuse_a=*/false, /*reuse_b=*/false);
        }

        // Store: channel = blk*4096 + i*64 + j; element = channel*Msz + Mv
        const int l3 = c_l3[t];
        const int gj = tj * 16 + lr;
        const long base = c_obase[l3] + (long)b * c_bstride[l3]
                        + (long)(c_blk[t] * 4096) * Msz + Mv;
        #pragma unroll
        for (int r = 0; r < 8; ++r) {
            const int i = ti * 16 + r + half * 8;   // C/D layout: VGPR r -> M=r / r+8
            out[base + (long)(i * 64 + gj) * Msz] = c[r];
        }
    }
}

extern "C" void kernel_launch(void* const* d_in, const int* in_sizes, int n_in,
                              void* d_out, int out_size, void* d_ws, size_t ws_size,
                              hipStream_t stream) {
    const float* x0 = (const float*)d_in[0];
    const float* x1 = (const float*)d_in[1];
    const float* x2 = (const float*)d_in[2];
    const int B = in_sizes[0] / 64;   // x0 is [B, 64, 1]
    tp_wmma_kernel<<<dim3(B), dim3(256), 0, stream>>>(
        x0, x1, x2,
        (const float*)d_in[3],  (const float*)d_in[4],  (const float*)d_in[5],
        (const float*)d_in[6],  (const float*)d_in[7],  (const float*)d_in[8],
        (const float*)d_in[9],  (const float*)d_in[10], (const float*)d_in[11],
        (const float*)d_in[12], (const float*)d_in[13],
        (float*)d_out, B);
}